// UltraSparseMemoryMLP_65240553226488
// MI455X (gfx1250) — compile-verified
//
#include <hip/hip_runtime.h>
#include <math.h>

typedef float f32x2 __attribute__((ext_vector_type(2)));
typedef float f32x8 __attribute__((ext_vector_type(8)));
typedef int   i32x4 __attribute__((ext_vector_type(4)));

#define NB    1024   // flattened batch b*s
#define NN    2048   // n
#define NR    16     // r
#define ND    16     // dq/(2r)
#define NK    256    // NR*ND
#define VSQ   2
#define VEF   4
#define NCORE 4
#define DVC   128
#define TK    16

#if defined(__gfx1250__) && __has_builtin(__builtin_amdgcn_global_load_async_to_lds_b128)
#define USM_ASYNC_LDS 1
#else
#define USM_ASYNC_LDS 0
#endif

// copy 16 bytes global -> LDS via the CDNA5 async-to-LDS path (ASYNCcnt-tracked);
// falls back to a b128 load + ds_store pair if the builtin is unavailable.
__device__ __forceinline__ void usm_copy16_g2l(const float* g, float* l) {
#if USM_ASYNC_LDS
  __builtin_amdgcn_global_load_async_to_lds_b128(
      (__attribute__((address_space(1))) i32x4*)g,
      (__attribute__((address_space(3))) i32x4*)l, 0, 0);
#else
  *(float4*)l = *(const float4*)g;
#endif
}
__device__ __forceinline__ void usm_wait_async() {
#if defined(__gfx1250__)
  asm volatile("s_wait_asynccnt 0x0" ::: "memory");
#endif
}

// ---------------- wave32 reductions ----------------
__device__ __forceinline__ float wave_sum32(float v) {
#pragma unroll
  for (int m = 16; m; m >>= 1) v += __shfl_xor(v, m, 32);
  return v;
}
__device__ __forceinline__ float wave_max32(float v) {
#pragma unroll
  for (int m = 16; m; m >>= 1) v = fmaxf(v, __shfl_xor(v, m, 32));
  return v;
}

__device__ __forceinline__ f32x8 wmma_f32_16x16x4(f32x2 a, f32x2 b, f32x8 c) {
  return __builtin_amdgcn_wmma_f32_16x16x4_f32(false, a, false, b, (short)0, c,
                                               false, false);
}

// ---------------- kernel 1: C = sum(Cs); one-sided Jacobi SVD ----------------
// ws layout (floats): [0..255]=C, [256..271]=t_vec (=Vh[:,0]), [272..287]=u_vec (=U[0,:])
__global__ void usm_svd_kernel(const float* __restrict__ Cs, float* __restrict__ ws) {
  if (threadIdx.x != 0 || blockIdx.x != 0) return;
  float a[NR][NR], v[NR][NR];
  for (int r = 0; r < NR; ++r)
    for (int c = 0; c < NR; ++c) {
      float s = 0.f;
      for (int k = 0; k < NCORE; ++k) s += Cs[k * 256 + r * 16 + c];
      a[r][c] = s;
      ws[r * 16 + c] = s;
      v[r][c] = (r == c) ? 1.f : 0.f;
    }
  // one-sided Jacobi: A <- A*J, V <- V*J ; at convergence A = U*Sigma
  for (int sweep = 0; sweep < 30; ++sweep) {
    for (int p = 0; p < NR - 1; ++p)
      for (int q = p + 1; q < NR; ++q) {
        float al = 0.f, be = 0.f, ga = 0.f;
        for (int k = 0; k < NR; ++k) {
          al += a[k][p] * a[k][p];
          be += a[k][q] * a[k][q];
          ga += a[k][p] * a[k][q];
        }
        if (fabsf(ga) < 1e-12f) continue;
        float zeta = (be - al) / (2.f * ga);
        float t = copysignf(1.f, zeta) / (fabsf(zeta) + sqrtf(1.f + zeta * zeta));
        float cc = rsqrtf(1.f + t * t);
        float ss = cc * t;
        for (int k = 0; k < NR; ++k) {
          float ap = a[k][p], aq = a[k][q];
          a[k][p] = cc * ap - ss * aq;
          a[k][q] = ss * ap + cc * aq;
          float vp = v[k][p], vq = v[k][q];
          v[k][p] = cc * vp - ss * vq;
          v[k][q] = ss * vp + cc * vq;
        }
      }
  }
  float sig[NR];
  int perm[NR];
  for (int j = 0; j < NR; ++j) {
    float s = 0.f;
    for (int k = 0; k < NR; ++k) s += a[k][j] * a[k][j];
    sig[j] = sqrtf(s);
    perm[j] = j;
  }
  for (int i = 0; i < NR - 1; ++i) {  // sort singular values descending
    int best = i;
    for (int j = i + 1; j < NR; ++j)
      if (sig[perm[j]] > sig[perm[best]]) best = j;
    int tmp = perm[i]; perm[i] = perm[best]; perm[best] = tmp;
  }
  for (int j = 0; j < NR; ++j) {
    int pj = perm[j];
    ws[256 + j] = v[0][pj];                               // t_vec[j] = Vh[j,0] = V[0,pj]
    ws[272 + j] = (sig[pj] > 0.f) ? a[0][pj] / sig[pj] : 0.f;  // u_vec[j] = U[0,j]
  }
}

// ---------------- kernel 2: pack A operand qa[kind][y][B][k]=w[x]*Qpart ----------------
__global__ void usm_pack_qa_kernel(const float* __restrict__ Q,
                                   const float* __restrict__ ws,
                                   float* __restrict__ qa) {
  unsigned idx = blockIdx.x * blockDim.x + threadIdx.x;  // 2^20 total
  int k = idx & 255;
  int B = (idx >> 8) & 1023;
  int y = (idx >> 18) & 1;
  int kind = (idx >> 19) & 1;  // 0=row (u_vec), 1=col (t_vec)
  int x = k >> 4, d = k & 15;
  float w = (kind == 0) ? ws[272 + x] : ws[256 + x];
  size_t qoff = (size_t)B * 1024 + (size_t)(x >> 3) * 512 + (kind ? 256 : 0) +
                (x & 7) * 32 + y * 16 + d;
  qa[idx] = w * Q[qoff];
}

// ---------------- kernel 3: pack B operand kb[kind][y][k][n]=K[x,n,y,d] ----------------
__global__ void usm_pack_kb_kernel(const float* __restrict__ K_row,
                                   const float* __restrict__ K_col,
                                   float* __restrict__ kb) {
  unsigned idx = blockIdx.x * blockDim.x + threadIdx.x;  // 2^21 total
  int n = idx & 2047;
  int k = (idx >> 11) & 255;
  int y = (idx >> 19) & 1;
  int kind = (idx >> 20) & 1;
  int x = k >> 4, d = k & 15;
  const float* src = kind ? K_col : K_row;
  kb[idx] = src[(size_t)x * 65536 + (size_t)n * 32 + y * 16 + d];
}

// ---------------- kernel 4: scores GEMM [1024x256]x[256x2048] via WMMA ----------------
// Block = 8 waves sharing one async-staged LDS B panel (256x16 f32 = 16 KB).
// scores[kind][B][y][n]; one 16x16 tile per wave32, EXEC all-ones.
__global__ void usm_scores_kernel(const float* __restrict__ qa,
                                  const float* __restrict__ kb,
                                  float* __restrict__ scores) {
  __shared__ float bpanel[NK * 16];  // [k][n-n0]
  int bg = blockIdx.x & 7;
  int nt = (blockIdx.x >> 3) & 127;
  int y = (blockIdx.x >> 10) & 1;
  int kind = (blockIdx.x >> 11) & 1;
  const float* A = qa + (size_t)(kind * 2 + y) * NB * NK;
  const float* Bm = kb + (size_t)(kind * 2 + y) * NK * NN;
  int n0 = nt * 16;

  // stage B panel: 1024 x b128 chunks; 4 consecutive lanes cover one 64B row
  {
    int t = threadIdx.x;
    int kr = t >> 2, c4 = t & 3;
#pragma unroll
    for (int i = 0; i < 4; ++i) {  // 4 rounds of 64 rows
      usm_copy16_g2l(Bm + (size_t)(kr + 64 * i) * NN + n0 + c4 * 4,
                     bpanel + (kr + 64 * i) * 16 + c4 * 4);
    }
    usm_wait_async();
  }
  __syncthreads();

  int wv = threadIdx.x >> 5;
  int lane = threadIdx.x & 31;
  int m = lane & 15;
  int kk = (lane >> 4) * 2;
  int b0 = (bg * 8 + wv) * 16;
  f32x8 acc = {};
#pragma unroll 4
  for (int k0 = 0; k0 < NK; k0 += 4) {
    f32x2 a, b;
    a.x = A[(size_t)(b0 + m) * NK + k0 + kk];
    a.y = A[(size_t)(b0 + m) * NK + k0 + kk + 1];
    b.x = bpanel[(k0 + kk) * 16 + m];
    b.y = bpanel[(k0 + kk + 1) * 16 + m];
    acc = wmma_f32_16x16x4(a, b, acc);
  }
  int mbase = (lane >> 4) * 8;
#pragma unroll
  for (int g = 0; g < 8; ++g) {
    int row = b0 + g + mbase;
    scores[(((size_t)kind * NB + row) * VSQ + y) * NN + n0 + m] = acc[g];
  }
}

// ---------------- kernel 5: top-16 indices per 2048-row (wave per row, LDS) ----------------
__global__ void usm_topk_kernel(const float* __restrict__ scores,
                                int* __restrict__ topi) {
  __shared__ float lds[4 * NN];  // 4 waves per 128-thread block
  unsigned wave = (blockIdx.x * blockDim.x + threadIdx.x) >> 5;  // 4096 rows
  int lane = threadIdx.x & 31;
  int slot = (threadIdx.x >> 5) & 3;
  const float* rowp = scores + (size_t)wave * NN;
  float* l = lds + slot * NN;
#pragma unroll
  for (int i = 0; i < NN / 32; ++i) l[lane + 32 * i] = rowp[lane + 32 * i];
  for (int sel = 0; sel < TK; ++sel) {
    float bv = -INFINITY;
    int bi = 1 << 30;
#pragma unroll
    for (int i = 0; i < NN / 32; ++i) {
      int n = lane + 32 * i;
      float vv = l[n];
      if (vv > bv || (vv == bv && n < bi)) { bv = vv; bi = n; }
    }
#pragma unroll
    for (int m = 16; m; m >>= 1) {
      float ov = __shfl_xor(bv, m, 32);
      int oi = __shfl_xor(bi, m, 32);
      if (ov > bv || (ov == bv && oi < bi)) { bv = ov; bi = oi; }
    }
    if (lane == 0) {
      topi[wave * TK + sel] = bi;
      l[bi] = -INFINITY;
    }
  }
}

// ---------------- kernel 6: sparse grid + stable softmax + top-16 probs ----------------
__global__ void usm_cand_kernel(const float* __restrict__ Q,
                                const float* __restrict__ K_row,
                                const float* __restrict__ K_col,
                                const float* __restrict__ ws,
                                const int* __restrict__ topi,
                                float* __restrict__ cs, int* __restrict__ ci) {
  unsigned wave = (blockIdx.x * blockDim.x + threadIdx.x) >> 5;  // 4096 = B*VEF
  int lane = threadIdx.x & 31;
  int p = wave & 3;
  int B = wave >> 2;
  int i = p >> 1, j = p & 1;
  int m = lane & 15;
  const int* colidx = topi + (((size_t)1 * NB + B) * VSQ + j) * TK;  // kind=1
  const int* rowidx = topi + (((size_t)0 * NB + B) * VSQ + i) * TK;  // kind=0
  int n = colidx[m];
  bool inrow = false;
#pragma unroll
  for (int t = 0; t < TK; ++t) inrow |= (rowidx[t] == n);

  float scol[NR];
#pragma unroll
  for (int z = 0; z < NR; ++z) {
    float s = 0.f;
    const float* kc = K_col + (size_t)z * 65536 + (size_t)n * 32 + j * 16;
    const float* qc = Q + (size_t)B * 1024 + (size_t)(z >> 3) * 512 + 256 + (z & 7) * 32 + j * 16;
#pragma unroll
    for (int d = 0; d < ND; ++d) s += kc[d] * qc[d];
    scol[z] = s;
  }
  float v = 0.f;
  if (inrow) {
    float srow[NR];
#pragma unroll
    for (int x = 0; x < NR; ++x) {
      float s = 0.f;
      const float* kr = K_row + (size_t)x * 65536 + (size_t)n * 32 + i * 16;
      const float* qr = Q + (size_t)B * 1024 + (size_t)(x >> 3) * 512 + (x & 7) * 32 + i * 16;
#pragma unroll
      for (int d = 0; d < ND; ++d) s += kr[d] * qr[d];
      srow[x] = s;
    }
#pragma unroll
    for (int x = 0; x < NR; ++x) {
      float cz = 0.f;
#pragma unroll
      for (int z = 0; z < NR; ++z) cz += ws[x * 16 + z] * scol[z];
      v += srow[x] * cz;
    }
  }
  // stable softmax over grid row: 16 candidates + 2032 implicit zeros
  float gm = wave_max32((lane < 16) ? v : -INFINITY);
  gm = fmaxf(gm, 0.f);
  float Z = wave_sum32((lane < 16) ? __expf(v - gm) : 0.f) +
            (float)(NN - TK) * __expf(-gm);
  float vv[TK]; int nn[TK];
#pragma unroll
  for (int t = 0; t < TK; ++t) { vv[t] = __shfl(v, t, 32); nn[t] = __shfl(n, t, 32); }
  if (lane == 0) {
    for (int a = 0; a < TK - 1; ++a) {  // sort desc, ties -> smaller index
      int best = a;
      for (int b = a + 1; b < TK; ++b)
        if (vv[b] > vv[best] || (vv[b] == vv[best] && nn[b] < nn[best])) best = b;
      float tv = vv[a]; vv[a] = vv[best]; vv[best] = tv;
      int tn = nn[a]; nn[a] = nn[best]; nn[best] = tn;
    }
    float outv[TK]; int outn[TK];
    int outc = 0;
    for (int t = 0; t < TK; ++t)
      if (vv[t] > 0.f) { outn[outc] = nn[t]; outv[outc] = __expf(vv[t] - gm) / Z; ++outc; }
    float zp = __expf(-gm) / Z;  // prob of zero-valued grid entries
    for (int nidx = 0; outc < TK; ++nidx) {
      bool skip = false;
      for (int c = 0; c < TK; ++c) skip |= (nn[c] == nidx && vv[c] != 0.f);
      if (!skip) { outn[outc] = nidx; outv[outc] = zp; ++outc; }
    }
    for (int t = 0; t < TK; ++t) {
      cs[((size_t)B * VEF + p) * TK + t] = outv[t];
      ci[((size_t)B * VEF + p) * TK + t] = outn[t];
    }
  }
}

// ---------------- kernel 7: NV[c][B][p*128+d] = sum_m phys[c,idx,d]*score ----------------
__global__ void usm_nv_kernel(const float* __restrict__ phys,
                              const float* __restrict__ cs,
                              const int* __restrict__ ci,
                              float* __restrict__ nv) {
  int d = threadIdx.x;
  unsigned bp = blockIdx.x;  // 16384 = c*B*p
  int p = bp & 3;
  int B = (bp >> 2) & 1023;
  int c = bp >> 12;
  float acc = 0.f;
#pragma unroll
  for (int m = 0; m < TK; ++m) {
    int n = ci[((size_t)B * VEF + p) * TK + m];
    float s = cs[((size_t)B * VEF + p) * TK + m];
    acc += phys[((size_t)c * NN + n) * DVC + d] * s;
  }
  nv[(((size_t)c * NB + B) * VEF + p) * DVC + d] = acc;
}

// ---------------- kernel 8: out[B][c*128+e] = NV[c][B][:512] . VP[c][:512][e] ----------------
// Block = 8 waves sharing one async-staged LDS VP panel (512x16 f32 = 32 KB).
__global__ void usm_proj_kernel(const float* __restrict__ nv,
                                const float* __restrict__ vp,
                                float* __restrict__ out) {
  __shared__ float vpanel[512 * 16];  // [k][e-n0]
  int bg = blockIdx.x & 7;
  int nt = (blockIdx.x >> 3) & 7;
  int c = blockIdx.x >> 6;
  const float* A = nv + (size_t)c * NB * 512;
  const float* Bm = vp + (size_t)c * 512 * DVC;  // VP is already [c][k][e]
  int n0 = nt * 16;

  {
    int t = threadIdx.x;
    int kr = t >> 2, c4 = t & 3;
#pragma unroll
    for (int i = 0; i < 8; ++i) {  // 8 rounds of 64 rows
      usm_copy16_g2l(Bm + (size_t)(kr + 64 * i) * DVC + n0 + c4 * 4,
                     vpanel + (kr + 64 * i) * 16 + c4 * 4);
    }
    usm_wait_async();
  }
  __syncthreads();

  int wv = threadIdx.x >> 5;
  int lane = threadIdx.x & 31;
  int m = lane & 15;
  int kk = (lane >> 4) * 2;
  int b0 = (bg * 8 + wv) * 16;
  f32x8 acc = {};
#pragma unroll 4
  for (int k0 = 0; k0 < 512; k0 += 4) {
    f32x2 a, b;
    a.x = A[(size_t)(b0 + m) * 512 + k0 + kk];
    a.y = A[(size_t)(b0 + m) * 512 + k0 + kk + 1];
    b.x = vpanel[(k0 + kk) * 16 + m];
    b.y = vpanel[(k0 + kk + 1) * 16 + m];
    acc = wmma_f32_16x16x4(a, b, acc);
  }
  int mbase = (lane >> 4) * 8;
#pragma unroll
  for (int g = 0; g < 8; ++g) {
    int row = b0 + g + mbase;
    out[(size_t)row * 512 + c * DVC + n0 + m] = acc[g];
  }
}

extern "C" void kernel_launch(void* const* d_in, const int* in_sizes, int n_in,
                              void* d_out, int out_size, void* d_ws, size_t ws_size,
                              hipStream_t stream) {
  const float* Q     = (const float*)d_in[0];
  const float* Cs    = (const float*)d_in[1];
  const float* K_row = (const float*)d_in[2];
  const float* K_col = (const float*)d_in[3];
  const float* phys  = (const float*)d_in[4];
  const float* vp    = (const float*)d_in[5];
  float* out = (float*)d_out;

  // workspace carve (floats): ~55.3 MB total
  float* wsf = (float*)d_ws;
  float* wSVD = wsf;                         // 512 (C + t_vec + u_vec)
  float* wQA  = wsf + 512;                   // 2*2*1024*256 = 1,048,576
  float* wKB  = wQA + (1u << 20);            // 2*2*256*2048 = 2,097,152
  float* wSC  = wKB + (1u << 21);            // 2*1024*2*2048 = 8,388,608
  int*   wTI  = (int*)(wSC + (1u << 23));    // 65,536 ints
  float* wCS  = (float*)(wTI + 65536);       // 65,536
  int*   wCI  = (int*)(wCS + 65536);         // 65,536
  float* wNV  = (float*)(wCI + 65536);       // 4*1024*512 = 2,097,152

  usm_svd_kernel<<<1, 32, 0, stream>>>(Cs, wSVD);
  usm_pack_qa_kernel<<<4096, 256, 0, stream>>>(Q, wSVD, wQA);
  usm_pack_kb_kernel<<<8192, 256, 0, stream>>>(K_row, K_col, wKB);
  usm_scores_kernel<<<4096, 256, 0, stream>>>(wQA, wKB, wSC);
  usm_topk_kernel<<<1024, 128, 0, stream>>>(wSC, wTI);
  usm_cand_kernel<<<1024, 128, 0, stream>>>(Q, K_row, K_col, wSVD, wTI, wCS, wCI);
  usm_nv_kernel<<<16384, 128, 0, stream>>>(phys, wCS, wCI, wNV);
  usm_proj_kernel<<<256, 256, 0, stream>>>(wNV, vp, out);
}